// masked_softmax_selected_6674379178454
// MI455X (gfx1250) — compile-verified
//
#include <hip/hip_runtime.h>

#define TOPK   32
#define MASKV  (-10000000.0f)
#define DDIM   4096
#define NT     256           // 8 waves of 32
#define VPT    16            // values per thread (4 x float4)

typedef int   v4i __attribute__((ext_vector_type(4)));
typedef float v4f __attribute__((ext_vector_type(4)));
typedef __attribute__((address_space(1))) v4i g_v4i;   // global int4
typedef __attribute__((address_space(3))) v4i l_v4i;   // LDS int4

__device__ __forceinline__ uint32_t sortable(float x) {
    uint32_t s = __float_as_uint(x);
    return s ^ ((uint32_t)((int32_t)s >> 31) | 0x80000000u);
}

__global__ __launch_bounds__(NT) void topk_softmax_rows(
        const float* __restrict__ X, float* __restrict__ Out) {
    __shared__ __align__(16) float rowbuf[DDIM];   // 16 KB staged row
    __shared__ int      hist[NT];                  // 256-bin radix histogram
    __shared__ int      s_wtot[8];                 // per-wave suffix totals
    __shared__ float    s_wred[8];                 // per-wave float reductions
    __shared__ uint32_t s_prefix;
    __shared__ int      s_k;

    const int tid  = threadIdx.x;
    const int lane = tid & 31;
    const int wave = tid >> 5;
    const long long row = blockIdx.x;
    const float* rowp = X + row * (long long)DDIM;
    float*       outp = Out + row * (long long)DDIM;

    // ---- Stage the row into LDS via CDNA5 async copies (ASYNCcnt path) ----
#if defined(__gfx1250__) && __has_builtin(__builtin_amdgcn_global_load_async_to_lds_b128)
    #pragma unroll
    for (int j = 0; j < 4; ++j) {
        const int c = tid + j * NT;                // 16-byte chunk index
        __builtin_amdgcn_global_load_async_to_lds_b128(
            (g_v4i*)(rowp + c * 4),
            (l_v4i*)(&rowbuf[c * 4]),
            0, 0);
    }
  #if __has_builtin(__builtin_amdgcn_s_wait_asynccnt)
    __builtin_amdgcn_s_wait_asynccnt(0);
  #else
    asm volatile("s_wait_asynccnt 0" ::: "memory");
  #endif
#else
    #pragma unroll
    for (int j = 0; j < 4; ++j) {
        const int c = tid + j * NT;
        *reinterpret_cast<float4*>(&rowbuf[c * 4]) =
            *reinterpret_cast<const float4*>(rowp + c * 4);
    }
#endif
    __syncthreads();

    // ---- Pull my 16 values into registers as (float, sortable-key) pairs ----
    uint32_t key[VPT];
    float    val[VPT];
    #pragma unroll
    for (int j = 0; j < 4; ++j) {
        const int c = tid + j * NT;
        const float4 v = *reinterpret_cast<const float4*>(&rowbuf[c * 4]);
        val[4*j+0] = v.x; val[4*j+1] = v.y; val[4*j+2] = v.z; val[4*j+3] = v.w;
        key[4*j+0] = sortable(v.x); key[4*j+1] = sortable(v.y);
        key[4*j+2] = sortable(v.z); key[4*j+3] = sortable(v.w);
    }

    // ---- Row max (wave32 shuffle reduce + cross-wave via LDS) ----
    float m = val[0];
    #pragma unroll
    for (int i = 1; i < VPT; ++i) m = fmaxf(m, val[i]);
    #pragma unroll
    for (int d = 16; d >= 1; d >>= 1) m = fmaxf(m, __shfl_down(m, d, 32));
    if (lane == 0) s_wred[wave] = m;
    __syncthreads();
    float rowmax = s_wred[0];
    #pragma unroll
    for (int w = 1; w < 8; ++w) rowmax = fmaxf(rowmax, s_wred[w]);
    // NOTE: every thread's s_wred reads happen before the first radix barrier;
    // s_wred is only rewritten after the final radix barrier -> no extra guard needed.

    // ---- 4-pass radix select: exact 32nd-largest sortable key ----
    uint32_t prefix = 0u;
    int      k      = TOPK;
    #pragma unroll
    for (int pass = 0; pass < 4; ++pass) {
        const int      shift = 24 - 8 * pass;
        const uint32_t pmask = (pass == 0) ? 0u : (0xFFFFFFFFu << (shift + 8));

        // hist[] last read before the s_wtot barrier of the previous pass,
        // so zeroing here (after the pass-end barrier) is race-free.
        hist[tid] = 0;
        __syncthreads();

        #pragma unroll
        for (int i = 0; i < VPT; ++i) {
            if ((key[i] & pmask) == prefix)
                atomicAdd(&hist[(key[i] >> shift) & 0xFF], 1);
        }
        __syncthreads();

        // two-level inclusive suffix sum over 256 bins (bin index == tid)
        const int h = hist[tid];
        int s = h;
        #pragma unroll
        for (int d = 1; d < 32; d <<= 1) {
            const int o = __shfl_down(s, d, 32);
            if (lane + d < 32) s += o;
        }
        if (lane == 0) s_wtot[wave] = s;   // wave total = suffix at lane 0
        __syncthreads();
        int above = 0;
        #pragma unroll
        for (int w = 0; w < 8; ++w) if (w > wave) above += s_wtot[w];
        const int g     = s + above;       // count of keys in bins >= tid (within prefix set)
        const int gnext = g - h;           // count of keys in bins >  tid
        if (g >= k && gnext < k) {         // exactly one bin satisfies this
            s_prefix = prefix | ((uint32_t)tid << shift);
            s_k      = k - gnext;
        }
        __syncthreads();
        prefix = s_prefix;
        k      = s_k;
    }
    const uint32_t T = prefix;             // bit pattern of the k-th largest value

    // ---- Masked exp + block sum ----
    float sum = 0.0f;
    #pragma unroll
    for (int i = 0; i < VPT; ++i) {
        const float xm = (key[i] >= T) ? val[i] : MASKV;  // >= keeps ties, like reference
        const float e  = __expf(xm - rowmax);             // masked lanes -> exactly 0
        val[i] = e;
        sum   += e;
    }
    #pragma unroll
    for (int d = 16; d >= 1; d >>= 1) sum += __shfl_down(sum, d, 32);
    if (lane == 0) s_wred[wave] = sum;     // last s_wred read was pre-radix -> safe
    __syncthreads();
    float tot = 0.0f;
    #pragma unroll
    for (int w = 0; w < 8; ++w) tot += s_wred[w];
    const float inv = 1.0f / tot;

    // ---- Store (coalesced b128, nontemporal: output is write-once stream) ----
    #pragma unroll
    for (int j = 0; j < 4; ++j) {
        const int c = tid + j * NT;
        v4f o;
        o.x = val[4*j+0] * inv; o.y = val[4*j+1] * inv;
        o.z = val[4*j+2] * inv; o.w = val[4*j+3] * inv;
        __builtin_nontemporal_store(o, (v4f*)(outp + c * 4));
    }
}

extern "C" void kernel_launch(void* const* d_in, const int* in_sizes, int n_in,
                              void* d_out, int out_size, void* d_ws, size_t ws_size,
                              hipStream_t stream) {
    const float* X   = (const float*)d_in[0];
    float*       Out = (float*)d_out;
    const int rows = in_sizes[0] / DDIM;   // 4*4096*4096 / 4096 = 16384 rows
    topk_softmax_rows<<<dim3(rows), dim3(NT), 0, stream>>>(X, Out);
}